// SQA_37984690766446
// MI455X (gfx1250) — compile-verified
//
#include <hip/hip_runtime.h>
#include <math.h>

// ---------------------------------------------------------------------------
// Shapes
// ---------------------------------------------------------------------------
#define DIM       2048
#define SEQ       8192
#define NUM_HEADS 16
#define HEAD_DIM  128       // DIM / NUM_HEADS
#define PROJ_DIM  512       // DIM / 4
#define LN_EPS    1e-5f

#define LCHUNKS   4         // split-K chunks for logits pass (K = DIM)
#define LK        (DIM / LCHUNKS)      // 512
#define YCHUNKS   8         // split-K chunks for y pass (K = SEQ)
#define YK        (SEQ / YCHUNKS)      // 1024

typedef __attribute__((ext_vector_type(2))) float v2f;
typedef __attribute__((ext_vector_type(4))) float v4f;
typedef __attribute__((ext_vector_type(8))) float v8f;

// ---------------------------------------------------------------------------
// K1: U[h, j] = sum_i q[h*128+i] * W_k[h*128+i, j]   (W_k = W_kv rows 0..2047)
//     c[h]    = sum_i q[h*128+i] * b_k[h*128+i]
// 32768 threads (128 blocks x 256).
// ---------------------------------------------------------------------------
__global__ void sqa_qproj(const float* __restrict__ q,
                          const float* __restrict__ Wkv,
                          const float* __restrict__ bkv,
                          float* __restrict__ U,
                          float* __restrict__ c) {
  const int tid = blockIdx.x * blockDim.x + threadIdx.x;  // 0..32767
  const int h = tid >> 11;        // / DIM
  const int j = tid & (DIM - 1);
  const float* qh = q + h * HEAD_DIM;
  float acc = 0.f;
  #pragma unroll 4
  for (int i = 0; i < HEAD_DIM; ++i)
    acc = fmaf(qh[i], Wkv[(size_t)(h * HEAD_DIM + i) * DIM + j], acc);
  U[h * DIM + j] = acc;
  if (j == 0) {
    const float* bh = bkv + h * HEAD_DIM;
    float cc = 0.f;
    for (int i = 0; i < HEAD_DIM; ++i) cc = fmaf(qh[i], bh[i], cc);
    c[h] = cc;
  }
}

// ---------------------------------------------------------------------------
// K2: partial logits via fp32 WMMA 16x16x4, split-K over LCHUNKS.
//   logitP[chunk][h][s] = sum_{k in chunk} x[s,k] * U[h,k]
// A layout (ISA 32-bit 16x4): lane<16 -> M=lane, K={k0,k0+1}; lane>=16 -> K={k0+2,k0+3}
// B layout (row-striped):     lane<16 -> N=lane, same K split.
// Grid: 512 blocks x 128 threads = 2048 waves (512 M-tiles x 4 chunks).
// ---------------------------------------------------------------------------
__global__ void sqa_logits_wmma(const float* __restrict__ x,
                                const float* __restrict__ U,
                                float* __restrict__ logitP) {
  const int lane  = threadIdx.x & 31;
  const int wave  = threadIdx.x >> 5;
  const int w     = blockIdx.x * (blockDim.x >> 5) + wave;  // 0..2047
  const int mtile = w >> 2;                  // 0..511
  const int chunk = w & (LCHUNKS - 1);       // 0..3
  const int mn    = lane & 15;               // M for A, N for B
  const int koff  = (lane >> 4) << 1;        // 0 or 2
  const int kbase = chunk * LK;
  const float* xrow = x + (size_t)(mtile * 16 + mn) * DIM + kbase + koff;
  const float* urow = U + (size_t)mn * DIM + kbase + koff;   // B[k][n] = U[n][k]
  v8f acc = {};
  #pragma unroll 4
  for (int k0 = 0; k0 < LK; k0 += 4) {
    v2f a; a.x = xrow[k0]; a.y = xrow[k0 + 1];
    v2f b; b.x = urow[k0]; b.y = urow[k0 + 1];
    acc = __builtin_amdgcn_wmma_f32_16x16x4_f32(false, a, false, b,
                                                (short)0, acc, false, false);
  }
  // D tile: VGPR r, lane l -> M = r + (l>=16 ? 8 : 0) (seq row), N = l&15 (head)
  const int head  = lane & 15;
  const int sbase = mtile * 16 + ((lane >> 4) << 3);
  float* dst = logitP + ((size_t)chunk * NUM_HEADS + head) * SEQ + sbase;
  #pragma unroll
  for (int r = 0; r < 8; ++r) dst[r] = acc[r];
}

// ---------------------------------------------------------------------------
// K3: combine logit partials + bias + scale, then softmax over s per head.
// 16 blocks x 256 threads.
// ---------------------------------------------------------------------------
__global__ void sqa_softmax(const float* __restrict__ logitP,
                            const float* __restrict__ c,
                            float* __restrict__ attnT) {
  __shared__ float red[256];
  const int h = blockIdx.x;
  const int t = threadIdx.x;
  const float bias  = c[h];
  const float scale = 0.08838834764831845f;   // 1/sqrt(128)
  float* row = attnT + (size_t)h * SEQ;
  const float* p = logitP + (size_t)h * SEQ;
  const size_t CS = (size_t)NUM_HEADS * SEQ;  // chunk stride
  // pass 1: combine partials, apply epilogue, track max
  float m = -INFINITY;
  for (int s = t; s < SEQ; s += 256) {
    float v = (p[s] + p[s + CS] + p[s + 2 * CS] + p[s + 3 * CS] + bias) * scale;
    row[s] = v;
    m = fmaxf(m, v);
  }
  red[t] = m; __syncthreads();
  for (int off = 128; off > 0; off >>= 1) {
    if (t < off) red[t] = fmaxf(red[t], red[t + off]);
    __syncthreads();
  }
  m = red[0]; __syncthreads();
  float sum = 0.f;
  for (int s = t; s < SEQ; s += 256) sum += expf(row[s] - m);
  red[t] = sum; __syncthreads();
  for (int off = 128; off > 0; off >>= 1) {
    if (t < off) red[t] += red[t + off];
    __syncthreads();
  }
  const float inv = 1.f / red[0];
  for (int s = t; s < SEQ; s += 256) row[s] = expf(row[s] - m) * inv;
}

// ---------------------------------------------------------------------------
// K4: partial y via fp32 WMMA 16x16x4, split-K over YCHUNKS.
//   yP[chunk][h][n] = sum_{s in chunk} attnT[h,s] * x[s,n]
// Grid: 256 blocks x 128 threads = 1024 waves (128 N-tiles x 8 chunks).
// ---------------------------------------------------------------------------
__global__ void sqa_yacc_wmma(const float* __restrict__ x,
                              const float* __restrict__ attnT,
                              float* __restrict__ yP) {
  const int lane  = threadIdx.x & 31;
  const int wave  = threadIdx.x >> 5;
  const int w     = blockIdx.x * (blockDim.x >> 5) + wave;  // 0..1023
  const int ntile = w >> 3;                  // 0..127
  const int chunk = w & (YCHUNKS - 1);       // 0..7
  const int mn    = lane & 15;
  const int koff  = (lane >> 4) << 1;
  const int col   = ntile * 16 + mn;
  const int kbase = chunk * YK;
  const float* arow = attnT + (size_t)mn * SEQ + kbase + koff;   // A: M = head
  const float* xcol = x + (size_t)kbase * DIM + col;             // B[k][n] = x[.][col]
  v8f acc = {};
  #pragma unroll 2
  for (int k0 = 0; k0 < YK; k0 += 4) {
    v2f a; a.x = arow[k0]; a.y = arow[k0 + 1];
    v2f b;
    b.x = xcol[(size_t)(k0 + koff) * DIM];
    b.y = xcol[(size_t)(k0 + koff + 1) * DIM];
    acc = __builtin_amdgcn_wmma_f32_16x16x4_f32(false, a, false, b,
                                                (short)0, acc, false, false);
  }
  const int hbase = (lane >> 4) << 3;   // D: M = r + (lane>=16 ? 8 : 0) = head
  float* dst = yP + (size_t)chunk * NUM_HEADS * DIM + col;
  #pragma unroll
  for (int r = 0; r < 8; ++r) dst[(size_t)(hbase + r) * DIM] = acc[r];
}

// ---------------------------------------------------------------------------
// K4b: y[i] = sum_c yP[c][i]   (i < 16*2048). 128 blocks x 256.
// ---------------------------------------------------------------------------
__global__ void sqa_reduce_y(const float* __restrict__ yP,
                             float* __restrict__ y) {
  const int i = blockIdx.x * blockDim.x + threadIdx.x;   // 0..32767
  float acc = 0.f;
  #pragma unroll
  for (int cck = 0; cck < YCHUNKS; ++cck)
    acc += yP[(size_t)cck * NUM_HEADS * DIM + i];
  y[i] = acc;
}

// ---------------------------------------------------------------------------
// K5a: attention output x1a[i] = W_v[i,:] . y[i/128, :] + b_v[i]  (i < 2048)
// ---------------------------------------------------------------------------
__global__ void sqa_attnout(const float* __restrict__ Wkv,
                            const float* __restrict__ bkv,
                            const float* __restrict__ y,
                            float* __restrict__ x1a) {
  __shared__ float red[256];
  const int i = blockIdx.x;           // 0..2047
  const int h = i >> 7;               // / HEAD_DIM
  const float* wv = Wkv + (size_t)(DIM + i) * DIM;   // W_v row
  const float* yh = y + (size_t)h * DIM;
  float acc = 0.f;
  for (int j = threadIdx.x; j < DIM; j += 256) acc = fmaf(wv[j], yh[j], acc);
  red[threadIdx.x] = acc; __syncthreads();
  for (int off = 128; off > 0; off >>= 1) {
    if (threadIdx.x < off) red[threadIdx.x] += red[threadIdx.x + off];
    __syncthreads();
  }
  if (threadIdx.x == 0) x1a[i] = red[0] + bkv[DIM + i];
}

// ---------------------------------------------------------------------------
// K5b: h1[p] = W_p1[p,:] . x1a + b_p1[p]   (p < 512)
// ---------------------------------------------------------------------------
__global__ void sqa_proj1(const float* __restrict__ Wp1,
                          const float* __restrict__ bp1,
                          const float* __restrict__ x1a,
                          float* __restrict__ h1) {
  __shared__ float red[256];
  const int p = blockIdx.x;           // 0..511
  const float* wp = Wp1 + (size_t)p * DIM;
  float acc = 0.f;
  for (int j = threadIdx.x; j < DIM; j += 256) acc = fmaf(wp[j], x1a[j], acc);
  red[threadIdx.x] = acc; __syncthreads();
  for (int off = 128; off > 0; off >>= 1) {
    if (threadIdx.x < off) red[threadIdx.x] += red[threadIdx.x + off];
    __syncthreads();
  }
  if (threadIdx.x == 0) h1[p] = red[0] + bp1[p];
}

// ---------------------------------------------------------------------------
// K5c: LayerNorm over 512 + affine + relu. Single block of 512 threads.
// ---------------------------------------------------------------------------
__global__ void sqa_ln_relu(const float* __restrict__ h1,
                            const float* __restrict__ lnw,
                            const float* __restrict__ lnb,
                            float* __restrict__ g) {
  __shared__ float red[PROJ_DIM];
  const int t = threadIdx.x;
  const float v = h1[t];
  red[t] = v; __syncthreads();
  for (int off = 256; off > 0; off >>= 1) {
    if (t < off) red[t] += red[t + off];
    __syncthreads();
  }
  const float mu = red[0] * (1.f / PROJ_DIM); __syncthreads();
  const float d = v - mu;
  red[t] = d * d; __syncthreads();
  for (int off = 256; off > 0; off >>= 1) {
    if (t < off) red[t] += red[t + off];
    __syncthreads();
  }
  const float var = red[0] * (1.f / PROJ_DIM);
  const float nv = d * rsqrtf(var + LN_EPS) * lnw[t] + lnb[t];
  g[t] = fmaxf(nv, 0.f);
}

// ---------------------------------------------------------------------------
// K5d: r[i] = W_p2[i,:] . g + b_p2[i]   (i < 2048)
// ---------------------------------------------------------------------------
__global__ void sqa_proj2(const float* __restrict__ Wp2,
                          const float* __restrict__ bp2,
                          const float* __restrict__ g,
                          float* __restrict__ r) {
  __shared__ float red[128];
  const int i = blockIdx.x;           // 0..2047
  const float* wp = Wp2 + (size_t)i * PROJ_DIM;
  float acc = 0.f;
  for (int p = threadIdx.x; p < PROJ_DIM; p += 128) acc = fmaf(wp[p], g[p], acc);
  red[threadIdx.x] = acc; __syncthreads();
  for (int off = 64; off > 0; off >>= 1) {
    if (threadIdx.x < off) red[threadIdx.x] += red[threadIdx.x + off];
    __syncthreads();
  }
  if (threadIdx.x == 0) r[i] = red[0] + bp2[i];
}

// ---------------------------------------------------------------------------
// K6: out[s, :] = x[s, :] + r[:].  Streaming: nontemporal stores keep the
// write-once output from evicting x/weights out of the 192MB L2.
// ---------------------------------------------------------------------------
__global__ void sqa_residual(const float* __restrict__ x,
                             const float* __restrict__ r,
                             float* __restrict__ out) {
  const size_t idx = (size_t)blockIdx.x * blockDim.x + threadIdx.x;
  const v4f* x4 = (const v4f*)x;
  const v4f* r4 = (const v4f*)r;
  v4f* o4 = (v4f*)out;
  const v4f a = x4[idx];
  const v4f b = r4[idx & (DIM / 4 - 1)];
  __builtin_nontemporal_store(a + b, &o4[idx]);
}

// ---------------------------------------------------------------------------
// Launch
// ---------------------------------------------------------------------------
extern "C" void kernel_launch(void* const* d_in, const int* in_sizes, int n_in,
                              void* d_out, int out_size, void* d_ws, size_t ws_size,
                              hipStream_t stream) {
  const float* x    = (const float*)d_in[0];
  const float* q    = (const float*)d_in[1];
  const float* Wkv  = (const float*)d_in[2];
  const float* bkv  = (const float*)d_in[3];
  const float* Wp1  = (const float*)d_in[4];
  const float* bp1  = (const float*)d_in[5];
  const float* Wp2  = (const float*)d_in[6];
  const float* bp2  = (const float*)d_in[7];
  const float* lnw  = (const float*)d_in[8];
  const float* lnb  = (const float*)d_in[9];
  float* out = (float*)d_out;

  // Workspace layout (floats) — ~4 MB total, all 16B-aligned slices.
  float* ws     = (float*)d_ws;
  float* U      = ws;                                    // 32768
  float* cvec   = U + NUM_HEADS * DIM;                   // 16 (pad to 32)
  float* attnT  = cvec + 32;                             // 131072
  float* logitP = attnT + (size_t)NUM_HEADS * SEQ;       // 4*16*8192 = 524288
  float* yP     = logitP + (size_t)LCHUNKS * NUM_HEADS * SEQ;  // 8*16*2048 = 262144
  float* y      = yP + (size_t)YCHUNKS * NUM_HEADS * DIM;      // 32768
  float* x1a    = y + NUM_HEADS * DIM;                   // 2048
  float* h1     = x1a + DIM;                             // 512
  float* g      = h1 + PROJ_DIM;                         // 512
  float* r      = g + PROJ_DIM;                          // 2048
  (void)ws_size; (void)in_sizes; (void)n_in; (void)out_size;

  sqa_qproj      <<<128, 256, 0, stream>>>(q, Wkv, bkv, U, cvec);
  sqa_logits_wmma<<<512, 128, 0, stream>>>(x, U, logitP);        // 2048 waves
  sqa_softmax    <<<NUM_HEADS, 256, 0, stream>>>(logitP, cvec, attnT);
  sqa_yacc_wmma  <<<256, 128, 0, stream>>>(x, attnT, yP);        // 1024 waves
  sqa_reduce_y   <<<128, 256, 0, stream>>>(yP, y);
  sqa_attnout    <<<DIM, 256, 0, stream>>>(Wkv, bkv, y, x1a);
  sqa_proj1      <<<PROJ_DIM, 256, 0, stream>>>(Wp1, bp1, x1a, h1);
  sqa_ln_relu    <<<1, PROJ_DIM, 0, stream>>>(h1, lnw, lnb, g);
  sqa_proj2      <<<DIM, 128, 0, stream>>>(Wp2, bp2, g, r);
  sqa_residual   <<<(SEQ * DIM / 4) / 256, 256, 0, stream>>>(x, r, out);
}